// LAT_79628693668197
// MI455X (gfx1250) — compile-verified
//
#include <hip/hip_runtime.h>
#include <math.h>

typedef __attribute__((ext_vector_type(2))) float v2f;
typedef __attribute__((ext_vector_type(8))) float v8f;

namespace {
constexpr int kB  = 16384;
constexpr int kV  = 8;
constexpr int kJ  = 21;
constexpr int kBJ = kB * kJ;                 // 344064 problems (b*J + j)
constexpr int kWavesPerBlock = 8;            // 256 threads, wave32
constexpr int kProbPerWave   = 4;            // 4 Gram blocks per 16x16 WMMA acc
constexpr int kBlockThreads  = kWavesPerBlock * 32;
constexpr int kProbPerBlock  = kWavesPerBlock * kProbPerWave;  // 32
constexpr int kGrid          = kBJ / kProbPerBlock;            // 10752 exact
}

// One wave = 4 DLT problems. Gram matrices G_q = A_qTA_q (4x4) are computed as
// the diagonal 4x4 blocks of D = sum_c Amat_c x Bmat_c using
// V_WMMA_F32_16X16X4_F32 (full f32 precision). Per the CDNA5 VGPR layouts
// (A-op: lane%16 = M, K = reg + 2*(lane>=16); B-op: lane%16 = N, same K rule),
// the block-diagonal A^T chunks and A chunks place *identical* values in each
// lane, so a single v2f per chunk feeds both operands.
__global__ __launch_bounds__(kBlockThreads)
void dlt_wmma_kernel(const float* __restrict__ Kin,   // (B,V,3,3) intrinsic
                     const float* __restrict__ Ein,   // (B,V,3,4) extrinsic
                     const float* __restrict__ UVin,  // (B,V,J,2)
                     const float* __restrict__ Cin,   // (B,V,J)
                     float* __restrict__ Out)         // (B,J,3)
{
  __shared__ float gram[kWavesPerBlock][256];

  const int tid  = threadIdx.x;
  const int lane = tid & 31;
  const int wv   = tid >> 5;
  const int p_base = (blockIdx.x * kWavesPerBlock + wv) * kProbPerWave;

  const int n    = lane & 15;     // N (and M) index within 16
  const int half = lane >> 4;     // selects K pair / M-high rows
  const int q    = n >> 2;        // problem slot 0..3 within wave
  const int col  = n & 3;         // column of A (0..3)

  const int p = p_base + q;       // flat problem id = b*kJ + j
  const int b = p / kJ;
  const int j = p - b * kJ;

  v8f acc = {0.f, 0.f, 0.f, 0.f, 0.f, 0.f, 0.f, 0.f};

  // K-chunks: chunk c covers A rows 4c..4c+3. For this lane both registers of
  // chunk c belong to view = 2c + half: reg.x = u-row, reg.y = v-row.
  #pragma unroll
  for (int c = 0; c < 4; ++c) {
    const int view = 2 * c + half;
    const int bv   = b * kV + view;
    // Column `col` of P = K(3x3) * E(3x4):  P[i][col] = sum_t K[i][t]*E[t][col]
    const float* Kp = Kin + bv * 9;
    const float* Ep = Ein + bv * 12 + col;
    const float e0 = Ep[0], e1 = Ep[4], e2 = Ep[8];
    const float P0 = Kp[0] * e0 + Kp[1] * e1 + Kp[2] * e2;
    const float P1 = Kp[3] * e0 + Kp[4] * e1 + Kp[5] * e2;
    const float P2 = Kp[6] * e0 + Kp[7] * e1 + Kp[8] * e2;

    const int bvj = bv * kJ + j;
    const float w = Cin[bvj];
    const float u = UVin[bvj * 2 + 0];
    const float v = UVin[bvj * 2 + 1];

    v2f ab;
    ab.x = w * (u * P2 - P0);   // A row 4c + 2*half     (u-row of this view)
    ab.y = w * (v * P2 - P1);   // A row 4c + 2*half + 1 (v-row of this view)

    // D += Amat x Bmat ; same registers serve as both operands (see above).
    acc = __builtin_amdgcn_wmma_f32_16x16x4_f32(
        /*neg_a=*/false, ab, /*neg_b=*/false, ab,
        /*c_mod=*/(short)0, acc, /*reuse_a=*/false, /*reuse_b=*/false);
  }

  // C/D layout: VGPR r, lane l -> M = r + 8*(l>=16), N = l%16.
  #pragma unroll
  for (int r = 0; r < 8; ++r) {
    gram[wv][(r + 8 * half) * 16 + n] = acc[r];
  }
  __syncthreads();

  // 4 solver lanes per wave: extract 4x4 Gram, null vector via adjugate.
  if (lane < 4) {
    const int pp = p_base + lane;
    const int bb = pp / kJ;
    const int jj = pp - bb * kJ;
    const float* G = &gram[wv][0];

    float a[4][4];
    float mx = 0.f;
    #pragma unroll
    for (int i = 0; i < 4; ++i) {
      #pragma unroll
      for (int k = 0; k < 4; ++k) {
        const float g = G[(4 * lane + i) * 16 + (4 * lane + k)];
        a[i][k] = g;
        mx = fmaxf(mx, fabsf(g));
      }
    }
    const float sc = (mx > 0.f) ? (1.0f / mx) : 0.f;
    #pragma unroll
    for (int i = 0; i < 4; ++i)
      #pragma unroll
      for (int k = 0; k < 4; ++k) a[i][k] *= sc;

    // Adjugate of 4x4 (glu-style 2x2 subdeterminants); G is PSD so
    // adj(G) ~ v4 v4^T for rank-3 G: any strong column is the null vector.
    const float s0 = a[0][0]*a[1][1] - a[1][0]*a[0][1];
    const float s1 = a[0][0]*a[1][2] - a[1][0]*a[0][2];
    const float s2 = a[0][0]*a[1][3] - a[1][0]*a[0][3];
    const float s3 = a[0][1]*a[1][2] - a[1][1]*a[0][2];
    const float s4 = a[0][1]*a[1][3] - a[1][1]*a[0][3];
    const float s5 = a[0][2]*a[1][3] - a[1][2]*a[0][3];
    const float c5 = a[2][2]*a[3][3] - a[3][2]*a[2][3];
    const float c4 = a[2][1]*a[3][3] - a[3][1]*a[2][3];
    const float c3 = a[2][1]*a[3][2] - a[3][1]*a[2][2];
    const float c2 = a[2][0]*a[3][3] - a[3][0]*a[2][3];
    const float c1 = a[2][0]*a[3][2] - a[3][0]*a[2][2];
    const float c0 = a[2][0]*a[3][1] - a[3][0]*a[2][1];

    float adj[4][4];
    adj[0][0] =  a[1][1]*c5 - a[1][2]*c4 + a[1][3]*c3;
    adj[0][1] = -a[0][1]*c5 + a[0][2]*c4 - a[0][3]*c3;
    adj[0][2] =  a[3][1]*s5 - a[3][2]*s4 + a[3][3]*s3;
    adj[0][3] = -a[2][1]*s5 + a[2][2]*s4 - a[2][3]*s3;
    adj[1][0] = -a[1][0]*c5 + a[1][2]*c2 - a[1][3]*c1;
    adj[1][1] =  a[0][0]*c5 - a[0][2]*c2 + a[0][3]*c1;
    adj[1][2] = -a[3][0]*s5 + a[3][2]*s2 - a[3][3]*s1;
    adj[1][3] =  a[2][0]*s5 - a[2][2]*s2 + a[2][3]*s1;
    adj[2][0] =  a[1][0]*c4 - a[1][1]*c2 + a[1][3]*c0;
    adj[2][1] = -a[0][0]*c4 + a[0][1]*c2 - a[0][3]*c0;
    adj[2][2] =  a[3][0]*s4 - a[3][1]*s2 + a[3][3]*s0;
    adj[2][3] = -a[2][0]*s4 + a[2][1]*s2 - a[2][3]*s0;
    adj[3][0] = -a[1][0]*c3 + a[1][1]*c1 - a[1][2]*c0;
    adj[3][1] =  a[0][0]*c3 - a[0][1]*c1 + a[0][2]*c0;
    adj[3][2] = -a[3][0]*s3 + a[3][1]*s1 - a[3][2]*s0;
    adj[3][3] =  a[2][0]*s3 - a[2][1]*s1 + a[2][2]*s0;

    // Pick column with largest (nonnegative) diagonal of adj.
    int jm = 0;
    float dm = adj[0][0];
    if (adj[1][1] > dm) { dm = adj[1][1]; jm = 1; }
    if (adj[2][2] > dm) { dm = adj[2][2]; jm = 2; }
    if (adj[3][3] > dm) { dm = adj[3][3]; jm = 3; }

    float v0[4];
    #pragma unroll
    for (int i = 0; i < 4; ++i) {
      v0[i] = (jm == 0) ? adj[i][0]
            : (jm == 1) ? adj[i][1]
            : (jm == 2) ? adj[i][2] : adj[i][3];
    }
    // One refinement pass: v1 = adj * v0 (adj is PSD -> no sign flip).
    float v1[4];
    #pragma unroll
    for (int i = 0; i < 4; ++i) {
      v1[i] = adj[i][0]*v0[0] + adj[i][1]*v0[1] + adj[i][2]*v0[2] + adj[i][3]*v0[3];
    }
    float n1 = v1[0]*v1[0] + v1[1]*v1[1] + v1[2]*v1[2] + v1[3]*v1[3];
    float w0 = v0[0], w1 = v0[1], w2 = v0[2], w3 = v0[3];
    if (n1 > 1e-30f) { w0 = v1[0]; w1 = v1[1]; w2 = v1[2]; w3 = v1[3]; }
    const float nn = w0*w0 + w1*w1 + w2*w2 + w3*w3;
    float x0 = 0.f, x1 = 0.f, x2 = 0.f, x3 = 0.f;
    if (nn > 1e-30f) {
      const float inv = 1.0f / sqrtf(nn);   // unit norm like SVD's Vh row
      x0 = w0 * inv; x1 = w1 * inv; x2 = w2 * inv; x3 = w3 * inv;
    }

    // Validity mask + dehomogenization (matches reference semantics).
    float tw = 0.f;
    const int base = bb * kV * kJ + jj;
    #pragma unroll
    for (int vw = 0; vw < kV; ++vw) tw += Cin[base + vw * kJ];
    const bool valid = (tw > 0.1f) && (fabsf(x3) > 1e-6f);
    float o0 = 0.f, o1 = 0.f, o2 = 0.f;
    if (valid) {
      const float invw = 1.0f / (x3 + 1e-8f);
      o0 = x0 * invw; o1 = x1 * invw; o2 = x2 * invw;
    }
    Out[pp * 3 + 0] = o0;
    Out[pp * 3 + 1] = o1;
    Out[pp * 3 + 2] = o2;
  }
}

extern "C" void kernel_launch(void* const* d_in, const int* in_sizes, int n_in,
                              void* d_out, int out_size, void* d_ws, size_t ws_size,
                              hipStream_t stream) {
  (void)in_sizes; (void)n_in; (void)d_ws; (void)ws_size; (void)out_size;
  const float* intrinsic   = (const float*)d_in[0];
  const float* extrinsic   = (const float*)d_in[1];
  const float* uv          = (const float*)d_in[2];
  const float* confidences = (const float*)d_in[3];
  float* out = (float*)d_out;

  dlt_wmma_kernel<<<kGrid, kBlockThreads, 0, stream>>>(
      intrinsic, extrinsic, uv, confidences, out);
}